// TGAM_75926431859194
// MI455X (gfx1250) — compile-verified
//
#include <hip/hip_runtime.h>

#define NPARTS 6
#define KNN    3
#define LSEQ   2048
#define CH     512
#define RATIO  (LSEQ / NPARTS)   // 341

typedef __attribute__((ext_vector_type(2))) float v2f;
typedef __attribute__((ext_vector_type(8))) float v8f;

// ---------------------------------------------------------------------------
// Kernel 1: streaming partial mean-pool. Dominates runtime (256 MiB read).
// grid = B * NPARTS * split blocks, 128 threads; thread t owns channels 4t..4t+3.
// Each row read is 2048 contiguous bytes per block (512 B per wave) -> coalesced.
// ---------------------------------------------------------------------------
__global__ void pool_partial(const float* __restrict__ x,
                             float* __restrict__ partials, int split) {
    int idx = blockIdx.x;
    int s = idx % split;
    int p = (idx / split) % NPARTS;
    int b = idx / (split * NPARTS);
    int r0 = p * RATIO + (RATIO * s) / split;
    int r1 = p * RATIO + (RATIO * (s + 1)) / split;
    const float4* base = (const float4*)(x + (size_t)b * LSEQ * CH);
    int t = threadIdx.x;  // 0..127
    float ax = 0.f, ay = 0.f, az = 0.f, aw = 0.f;
    int r = r0;
    // 4-deep unroll: 4 independent 16B loads in flight per thread.
    for (; r + 4 <= r1; r += 4) {
        float4 a0 = base[(size_t)(r + 0) * (CH / 4) + t];
        float4 a1 = base[(size_t)(r + 1) * (CH / 4) + t];
        float4 a2 = base[(size_t)(r + 2) * (CH / 4) + t];
        float4 a3 = base[(size_t)(r + 3) * (CH / 4) + t];
        ax += a0.x + a1.x + a2.x + a3.x;
        ay += a0.y + a1.y + a2.y + a3.y;
        az += a0.z + a1.z + a2.z + a3.z;
        aw += a0.w + a1.w + a2.w + a3.w;
    }
    for (; r < r1; ++r) {
        float4 a0 = base[(size_t)r * (CH / 4) + t];
        ax += a0.x; ay += a0.y; az += a0.z; aw += a0.w;
    }
    float4 out; out.x = ax; out.y = ay; out.z = az; out.w = aw;
    ((float4*)partials)[(size_t)idx * (CH / 4) + t] = out;
}

// ---------------------------------------------------------------------------
// Kernel 2: reduce split partials -> parts[B*NPARTS][CH], scale by 1/RATIO.
// ---------------------------------------------------------------------------
__global__ void pool_reduce(const float* __restrict__ partials,
                            float* __restrict__ parts, int split) {
    int g = blockIdx.x;      // 0 .. B*NPARTS-1
    int t = threadIdx.x;     // 0..127
    float ax = 0.f, ay = 0.f, az = 0.f, aw = 0.f;
    for (int s = 0; s < split; ++s) {
        float4 a = ((const float4*)partials)[((size_t)g * split + s) * (CH / 4) + t];
        ax += a.x; ay += a.y; az += a.z; aw += a.w;
    }
    const float inv = 1.0f / (float)RATIO;
    float4 out; out.x = ax * inv; out.y = ay * inv; out.z = az * inv; out.w = aw * inv;
    ((float4*)parts)[(size_t)g * (CH / 4) + t] = out;
}

// ---------------------------------------------------------------------------
// Kernel 3: per-batch graph: pairwise dist^2, top-3 (strict < matches top_k
// lowest-index tie-break; sqrt is monotonic so dist^2 ordering is identical),
// adjacency row-sums are always exactly KNN, so norm = 1/(KNN+1e-6).
// Outputs M[b][c] = mean_n(agg), Pbar[b][c] = mean_n(parts).
// ---------------------------------------------------------------------------
__global__ void graph_agg(const float* __restrict__ parts,
                          float* __restrict__ Mmat, float* __restrict__ Pbar) {
    __shared__ float sp[NPARTS][CH];
    __shared__ float sq[NPARTS][NPARTS];
    __shared__ int   nbr[NPARTS];
    __shared__ float cnt[NPARTS];
    int b = blockIdx.x, t = threadIdx.x;  // 256 threads = 8 waves
    for (int i = t; i < NPARTS * CH; i += 256)
        sp[i / CH][i % CH] = parts[(size_t)b * NPARTS * CH + i];
    if (t < NPARTS) sq[t][t] = 0.f;
    __syncthreads();

    const int pi[15] = {0,0,0,0,0,1,1,1,1,2,2,2,3,3,4};
    const int pj[15] = {1,2,3,4,5,2,3,4,5,3,4,5,4,5,5};
    int w = t >> 5, lane = t & 31;
    for (int p = w; p < 15; p += 8) {
        int i = pi[p], j = pj[p];
        float acc = 0.f;
        for (int c = lane; c < CH; c += 32) {
            float d = sp[i][c] - sp[j][c];
            acc += d * d;
        }
        for (int off = 16; off > 0; off >>= 1)
            acc += __shfl_down(acc, off, 32);
        if (lane == 0) { sq[i][j] = acc; sq[j][i] = acc; }
    }
    __syncthreads();

    if (t < NPARTS) {
        int i = t;
        bool used[NPARTS] = {false, false, false, false, false, false};
        int mask = 0;
        for (int k = 0; k < KNN; ++k) {
            int best = -1;
            for (int j = 0; j < NPARTS; ++j)
                if (!used[j] && (best < 0 || sq[i][j] < sq[i][best])) best = j;
            used[best] = true;
            mask |= (1 << best);
        }
        nbr[i] = mask;
    }
    __syncthreads();
    if (t == 0) {
        for (int j = 0; j < NPARTS; ++j) {
            int c = 0;
            for (int i = 0; i < NPARTS; ++i) c += (nbr[i] >> j) & 1;
            cnt[j] = (float)c;
        }
    }
    __syncthreads();

    const float coefA = (1.0f / (float)NPARTS) * (1.0f / ((float)KNN + 1e-6f));
    const float coefP = 1.0f / (float)NPARTS;
    for (int c = t; c < CH; c += 256) {
        float m = 0.f, pm = 0.f;
        for (int j = 0; j < NPARTS; ++j) {
            m  += cnt[j] * sp[j][c];
            pm += sp[j][c];
        }
        Mmat[(size_t)b * CH + c] = m * coefA;
        Pbar[(size_t)b * CH + c] = pm * coefP;
    }
}

// ---------------------------------------------------------------------------
// Kernel 4: out[B,CH] = M @ W^T + bias + Pbar via V_WMMA_F32_16X16X4_F32.
// One wave per 16x16 tile; EXEC all-ones (32-thread blocks, no divergence).
// A frag (16x4, ISA layout): v0 = K=0|K=2 across lane halves, v1 = K=1|K=3.
// B frag (4x16, row-striped): v0 = rows K=0|K=1, v1 = rows K=2|K=3.
// D layout: VGPR r -> M = r (+8 for lanes 16-31), N = lane&15.
// ---------------------------------------------------------------------------
__global__ void gemm_wmma(const float* __restrict__ Mmat,
                          const float* __restrict__ W,
                          const float* __restrict__ bias,
                          const float* __restrict__ Pbar,
                          float* __restrict__ out) {
    int nt = blockIdx.x % (CH / 16);
    int mt = blockIdx.x / (CH / 16);
    int lane = threadIdx.x;        // 0..31
    int half = lane >> 4;
    int l15  = lane & 15;
    int m0 = mt * 16, n0 = nt * 16;

    const float* Arow = Mmat + (size_t)(m0 + l15) * CH;  // A row for this lane
    const float* Bcol = W    + (size_t)(n0 + l15) * CH;  // W[d=n][c=k] row = B column

    v8f acc = {};
    for (int k = 0; k < CH; k += 4) {
        v2f a, bm;
        a.x  = Arow[k + 2 * half];       // K = 0 / 2
        a.y  = Arow[k + 2 * half + 1];   // K = 1 / 3
        bm.x = Bcol[k + half];           // rows K = 0 / 1
        bm.y = Bcol[k + 2 + half];       // rows K = 2 / 3
        acc = __builtin_amdgcn_wmma_f32_16x16x4_f32(
            /*neg_a=*/false, a, /*neg_b=*/false, bm,
            /*c_mod=*/(short)0, acc, /*reuse_a=*/false, /*reuse_b=*/false);
    }

    int n = n0 + l15;
    float bn = bias[n];
#pragma unroll
    for (int r = 0; r < 8; ++r) {
        int m = m0 + r + 8 * half;
        out[(size_t)m * CH + n] = acc[r] + bn + Pbar[(size_t)m * CH + n];
    }
}

// ---------------------------------------------------------------------------
extern "C" void kernel_launch(void* const* d_in, const int* in_sizes, int n_in,
                              void* d_out, int out_size, void* d_ws, size_t ws_size,
                              hipStream_t stream) {
    const float* x    = (const float*)d_in[0];
    const float* W    = (const float*)d_in[1];
    const float* bias = (const float*)d_in[2];
    float* out = (float*)d_out;

    int B = in_sizes[0] / (LSEQ * CH);   // 64

    // Pick row-split for kernel 1 based on available workspace (deterministic).
    int split = 8;
    while (split > 1) {
        size_t need = ((size_t)B * NPARTS * split * CH   // partials
                     + (size_t)B * NPARTS * CH           // parts
                     + 2 * (size_t)B * CH) * sizeof(float);
        if (need <= ws_size) break;
        split >>= 1;
    }

    float* partials = (float*)d_ws;
    float* parts    = partials + (size_t)B * NPARTS * split * CH;
    float* Mmat     = parts    + (size_t)B * NPARTS * CH;
    float* Pbar     = Mmat     + (size_t)B * CH;

    pool_partial<<<B * NPARTS * split, 128, 0, stream>>>(x, partials, split);
    pool_reduce <<<B * NPARTS,         128, 0, stream>>>(partials, parts, split);
    graph_agg   <<<B,                  256, 0, stream>>>(parts, Mmat, Pbar);
    gemm_wmma   <<<(B / 16) * (CH / 16), 32, 0, stream>>>(Mmat, W, bias, Pbar, out);
}